// HamiltonianFFN_33363305955952
// MI455X (gfx1250) — compile-verified
//
#include <hip/hip_runtime.h>
#include <hip/hip_bf16.h>
#include <math.h>

// HamiltonianFFN leapfrog on MI455X (gfx1250): one wave32 per particle,
// all 16x16 fp32 matrix products via V_WMMA_F32_16X16X4_F32 (4 chunks of K=4),
// matrices staged in LDS as 16x17-padded tiles (64-bank conflict free).
// eigh-based SPD functions replaced by matmul-only iterations:
//   - Newton-Schulz (trace-normalized, ping-pong buffers) for Sigma^{+-1/2}
//   - scaling-and-squaring (even s, ping-pong) + order-6 Taylor for exp(W)
//   - ensure_spd ~= sym(.) + eps*I

typedef __attribute__((ext_vector_type(2))) float v2f;
typedef __attribute__((ext_vector_type(8))) float v8f;

#define KD 16
#define LDP 17          // padded row stride (dwords) -> conflict-free on 64 banks
#define DT_C 0.05f
#define EPS_C 1e-8f
#define NSTEPS 4
#define NS_ITERS 12
#define WPB 4           // waves (particles) per block

// ---------------------------------------------------------------------------
// 16x16x16 fp32 matmul: D = scale * (X @ Y), X,Y,D are LDS tiles [16][LDP].
// Uses 4x V_WMMA_F32_16X16X4_F32.
// f32 A 16x4 layout (ISA 7.12.2): VGPR j, lane-half h: element (m = lane&15, K = k0+2h+j)
// f32 B 4x16 layout (row-striped, halves split across lane groups): VGPR j, half h:
//   element (K = k0+2h+j, n = lane&15)
// f32 C/D 16x16 layout: VGPR r, half h: element (row = r+8h, col = lane&15)
// ---------------------------------------------------------------------------
__device__ __forceinline__ void mm_wmma(const float* __restrict__ X,
                                        const float* __restrict__ Y,
                                        float* __restrict__ D,
                                        float scale, int lane) {
  const int m = lane & 15;
  const int h = lane >> 4;
  v8f acc = {};
#pragma unroll
  for (int k0 = 0; k0 < KD; k0 += 4) {
    const int ka = k0 + 2 * h;
    v2f a = { X[m * LDP + ka], X[m * LDP + ka + 1] };
    v2f b = { Y[ka * LDP + m], Y[(ka + 1) * LDP + m] };
    acc = __builtin_amdgcn_wmma_f32_16x16x4_f32(
        /*neg_a=*/false, a, /*neg_b=*/false, b,
        /*c_mod=*/(short)0, acc, /*reuse_a=*/false, /*reuse_b=*/false);
  }
#pragma unroll
  for (int r = 0; r < 8; ++r)
    D[(r + 8 * h) * LDP + m] = scale * acc[r];
}

// PS = sym(PS - dt * sym(T))   (gather-then-scatter: safe in-place within a wave)
__device__ __forceinline__ void kick_update(float* __restrict__ PS,
                                            const float* __restrict__ T, int lane) {
  float nv[8];
#pragma unroll
  for (int it = 0; it < 8; ++it) {
    const int idx = it * 32 + lane;
    const int r = idx >> 4, c = idx & 15;
    const float st = 0.5f * (T[r * LDP + c] + T[c * LDP + r]);
    nv[it] = 0.5f * (PS[r * LDP + c] + PS[c * LDP + r]) - DT_C * st;
  }
#pragma unroll
  for (int it = 0; it < 8; ++it) {
    const int idx = it * 32 + lane;
    const int r = idx >> 4, c = idx & 15;
    PS[r * LDP + c] = nv[it];
  }
}

// SG = sym(SG) + eps*I  (in-place, two-phase)
__device__ __forceinline__ void symeps_inplace(float* __restrict__ SG, int lane) {
  float nv[8];
#pragma unroll
  for (int it = 0; it < 8; ++it) {
    const int idx = it * 32 + lane;
    const int r = idx >> 4, c = idx & 15;
    nv[it] = 0.5f * (SG[r * LDP + c] + SG[c * LDP + r]) + ((r == c) ? EPS_C : 0.0f);
  }
#pragma unroll
  for (int it = 0; it < 8; ++it) {
    const int idx = it * 32 + lane;
    const int r = idx >> 4, c = idx & 15;
    SG[r * LDP + c] = nv[it];
  }
}

__global__ void __launch_bounds__(32 * WPB, 1)
hamiltonian_leapfrog(const float* __restrict__ g_mu, const float* __restrict__ g_Sigma,
                     const float* __restrict__ g_phi, const float* __restrict__ g_pimu,
                     const float* __restrict__ g_piS, const float* __restrict__ g_piphi,
                     const float* __restrict__ g_Minv,
                     float* __restrict__ o_mu, float* __restrict__ o_Sigma,
                     float* __restrict__ o_phi, int P) {
  __shared__ float lds[WPB][9][KD * LDP];
  const int lane = threadIdx.x & 31;
  const int w = threadIdx.x >> 5;
  const int p = blockIdx.x * WPB + w;
  if (p >= P) return;  // wave-uniform: EXEC stays all-ones for live waves

  float* SG = lds[w][0];  // Sigma
  float* PS = lds[w][1];  // pi_Sigma
  float* T0 = lds[w][2];
  float* T1 = lds[w][3];
  float* Y0 = lds[w][4];  // Newton-Schulz Y ping-pong
  float* Y1 = lds[w][5];
  float* Z0 = lds[w][6];  // Newton-Schulz Z ping-pong
  float* Z1 = lds[w][7];
  float* EM = lds[w][8];  // V, then W, then matmul temp

  const size_t mbase = (size_t)p * 256;

  // ---- load Sigma, pi_Sigma (coalesced 128B per iter per wave) ----
#pragma unroll
  for (int it = 0; it < 8; ++it) {
    const int idx = it * 32 + lane;
    const int r = idx >> 4, c = idx & 15;
    SG[r * LDP + c] = g_Sigma[mbase + idx];
    PS[r * LDP + c] = g_piS[mbase + idx];
  }

  // ---- mu: pi_mu is never kicked => mu_out = mu + NSTEPS*dt * Minv@pi_mu ----
  if (lane < KD) {
    float acc = 0.0f;
#pragma unroll
    for (int j = 0; j < KD; ++j)
      acc += g_Minv[mbase + (size_t)lane * KD + j] * g_pimu[(size_t)p * KD + j];
    o_mu[(size_t)p * KD + lane] =
        g_mu[(size_t)p * KD + lane] + ((float)NSTEPS * DT_C) * acc;
  }

  // ---- phi held redundantly per lane (uniform broadcast loads) ----
  float ph0 = g_phi[(size_t)p * 3 + 0];
  float ph1 = g_phi[(size_t)p * 3 + 1];
  float ph2 = g_phi[(size_t)p * 3 + 2];
  const float pp0 = g_piphi[(size_t)p * 3 + 0];
  const float pp1 = g_piphi[(size_t)p * 3 + 1];
  const float pp2 = g_piphi[(size_t)p * 3 + 2];

  const float PI_F = 3.14159265358979f;
  const float TWO_PI = 6.28318530717959f;

  for (int step = 0; step < NSTEPS; ++step) {
    // ===== half kick: PS = sym(PS - dt*sym(PS@SG@PS)) =====
    mm_wmma(PS, SG, T0, 1.0f, lane);
    mm_wmma(T0, PS, T1, 1.0f, lane);
    kick_update(PS, T1, lane);

    // ===== V = dt*Sigma_dot = 2dt * SG@PS@SG  -> EM =====
    mm_wmma(SG, PS, T0, 1.0f, lane);
    mm_wmma(T0, SG, EM, 2.0f * DT_C, lane);

    // ===== exp map: Sigma' = sym(SG)+eps*I; trace-normalized Newton-Schulz =====
    symeps_inplace(SG, lane);

    float tr = 0.0f;
#pragma unroll
    for (int r = 0; r < KD; ++r) tr += SG[r * LDP + r];  // broadcast reads, uniform
    const float rinv = 1.0f / tr;

#pragma unroll
    for (int it = 0; it < 8; ++it) {
      const int idx = it * 32 + lane;
      const int r = idx >> 4, c = idx & 15;
      Y0[r * LDP + c] = SG[r * LDP + c] * rinv;
      Z0[r * LDP + c] = (r == c) ? 1.0f : 0.0f;
    }

    // Newton-Schulz with ping-pong buffers (no copy passes)
    float* Ya = Y0; float* Yb = Y1;
    float* Za = Z0; float* Zb = Z1;
    for (int ns = 0; ns < NS_ITERS; ++ns) {
      mm_wmma(Za, Ya, T0, 1.0f, lane);
      // T0 = 1.5I - 0.5*T0  (same-element read/write: safe)
#pragma unroll
      for (int it = 0; it < 8; ++it) {
        const int idx = it * 32 + lane;
        const int r = idx >> 4, c = idx & 15;
        const float t = T0[r * LDP + c];
        T0[r * LDP + c] = ((r == c) ? 1.5f : 0.0f) - 0.5f * t;
      }
      mm_wmma(Ya, T0, Yb, 1.0f, lane);
      mm_wmma(T0, Za, Zb, 1.0f, lane);
      float* tmp = Ya; Ya = Yb; Yb = tmp;
      tmp = Za; Za = Zb; Zb = tmp;
    }
    // Ya ~ A^{1/2}, Za ~ A^{-1/2}, A = SG/tr => S^{1/2}=sqrt(tr)Ya, S^{-1/2}=Za/sqrt(tr)

    // ===== W = sym(S^{-1/2} V S^{-1/2}) = sym(Za@V@Za)/tr  -> EM =====
    mm_wmma(Za, EM, T0, 1.0f, lane);
    mm_wmma(T0, Za, T1, 1.0f, lane);
#pragma unroll
    for (int it = 0; it < 8; ++it) {
      const int idx = it * 32 + lane;
      const int r = idx >> 4, c = idx & 15;
      EM[r * LDP + c] = 0.5f * (T1[r * LDP + c] + T1[c * LDP + r]) * rinv;
    }

    // ===== exp(W): scaling-and-squaring (even s) + order-6 Taylor =====
    float fn2 = 0.0f;
#pragma unroll
    for (int it = 0; it < 8; ++it) {
      const int idx = it * 32 + lane;
      const int r = idx >> 4, c = idx & 15;
      const float v = EM[r * LDP + c];
      fn2 += v * v;
    }
#pragma unroll
    for (int msk = 16; msk >= 1; msk >>= 1) fn2 += __shfl_xor(fn2, msk, 32);
    const float fn = sqrtf(fn2);            // wave-uniform
    int s = 0;
    if (fn > 0.25f) {
      s = (int)ceilf(log2f(fn)) + 2;        // ||W/2^s|| <~ 0.25
      if (s < 0) s = 0;
      if (s > 14) s = 14;
    }
    s = (s + 1) & ~1;                       // round up to even: 2 squarings/trip
    const float sc = exp2f(-(float)s);

    // T1 = X = W*2^-s ; SG = E = I + X ; T0 = X (running term X^k/k!)
#pragma unroll
    for (int it = 0; it < 8; ++it) {
      const int idx = it * 32 + lane;
      const int r = idx >> 4, c = idx & 15;
      const float x = EM[r * LDP + c] * sc;
      T1[r * LDP + c] = x;
      T0[r * LDP + c] = x;
      SG[r * LDP + c] = ((r == c) ? 1.0f : 0.0f) + x;
    }
    {
      float* Ta = T0; float* Tb = EM;       // term ping-pong
      for (int k = 2; k <= 6; ++k) {
        mm_wmma(Ta, T1, Tb, 1.0f / (float)k, lane);
#pragma unroll
        for (int it = 0; it < 8; ++it) {
          const int idx = it * 32 + lane;
          const int r = idx >> 4, c = idx & 15;
          SG[r * LDP + c] += Tb[r * LDP + c];
        }
        float* tmp = Ta; Ta = Tb; Tb = tmp;
      }
    }
    for (int i = 0; i < s; i += 2) {        // wave-uniform trip count, no copies
      mm_wmma(SG, SG, EM, 1.0f, lane);
      mm_wmma(EM, EM, SG, 1.0f, lane);
    }

    // ===== Sigma_new = ensure_spd(S^{1/2} E S^{1/2}) ~= tr*sym(Ya@E@Ya) + eps*I =====
    mm_wmma(Ya, SG, T0, 1.0f, lane);
    mm_wmma(T0, Ya, T1, tr, lane);
#pragma unroll
    for (int it = 0; it < 8; ++it) {
      const int idx = it * 32 + lane;
      const int r = idx >> 4, c = idx & 15;
      SG[r * LDP + c] =
          0.5f * (T1[r * LDP + c] + T1[c * LDP + r]) + ((r == c) ? EPS_C : 0.0f);
    }

    // ===== phi retraction (scalar, redundant per lane) =====
    {
      ph0 += DT_C * pp0; ph1 += DT_C * pp1; ph2 += DT_C * pp2;
      const float theta = sqrtf(ph0 * ph0 + ph1 * ph1 + ph2 * ph2);
      const float ts = fmaxf(theta, 1e-12f);
      float ax0 = ph0 / ts, ax1 = ph1 / ts, ax2 = ph2 / ts;
      float tw = fmodf(theta, TWO_PI);
      const bool flip = tw > PI_F;
      float tf = flip ? (TWO_PI - tw) : tw;
      const float sg = flip ? -1.0f : 1.0f;
      tf = fminf(tf, PI_F - 0.01f);
      ph0 = sg * ax0 * tf; ph1 = sg * ax1 * tf; ph2 = sg * ax2 * tf;
    }

    // ===== second half kick (with new Sigma) =====
    mm_wmma(PS, SG, T0, 1.0f, lane);
    mm_wmma(T0, PS, T1, 1.0f, lane);
    kick_update(PS, T1, lane);
  }

  // ---- store Sigma (coalesced) and phi ----
#pragma unroll
  for (int it = 0; it < 8; ++it) {
    const int idx = it * 32 + lane;
    const int r = idx >> 4, c = idx & 15;
    o_Sigma[mbase + idx] = SG[r * LDP + c];
  }
  if (lane == 0) {
    o_phi[(size_t)p * 3 + 0] = ph0;
    o_phi[(size_t)p * 3 + 1] = ph1;
    o_phi[(size_t)p * 3 + 2] = ph2;
  }
}

extern "C" void kernel_launch(void* const* d_in, const int* in_sizes, int n_in,
                              void* d_out, int out_size, void* d_ws, size_t ws_size,
                              hipStream_t stream) {
  (void)n_in; (void)d_ws; (void)ws_size; (void)out_size;
  const float* g_mu    = (const float*)d_in[0];
  const float* g_Sigma = (const float*)d_in[1];
  const float* g_phi   = (const float*)d_in[2];
  const float* g_pimu  = (const float*)d_in[3];
  const float* g_piS   = (const float*)d_in[4];
  const float* g_piphi = (const float*)d_in[5];
  const float* g_Minv  = (const float*)d_in[6];

  const int P = in_sizes[0] / KD;  // B*N particles

  float* out = (float*)d_out;
  float* o_mu    = out;
  float* o_Sigma = out + (size_t)P * KD;
  float* o_phi   = out + (size_t)P * KD + (size_t)P * KD * KD;

  dim3 block(32 * WPB);
  dim3 grid((P + WPB - 1) / WPB);
  hamiltonian_leapfrog<<<grid, block, 0, stream>>>(
      g_mu, g_Sigma, g_phi, g_pimu, g_piS, g_piphi, g_Minv,
      o_mu, o_Sigma, o_phi, P);
}